// AttentionSiphon_31473520345697
// MI455X (gfx1250) — compile-verified
//
#include <hip/hip_runtime.h>
#include <math.h>

// Problem constants (from reference)
#define BD   4
#define NN   2046
#define IND  512
#define DD   1024
#define HH   16
#define DHD  64
#define SSQ  2048
#define EPSV 1e-5f

typedef float v2f __attribute__((ext_vector_type(2)));
typedef float v4f __attribute__((ext_vector_type(4)));
typedef float v8f __attribute__((ext_vector_type(8)));

// Workspace layout (float offsets)
//  [0,2048)              gw[s][d]   : ln_g * (Wk[h] @ q[s,h]) / 8
//  [2048,2080)           A[s][h]    : sum_{d in head} gw
//  [2080,2112)           C[s][h]    : sum_d ln_b*wv + (q.bk)/8
//  [4096,4096+262144)    scores[b][s][h][t]              (1 MB)
//  [266240,266240+524288) Bp: W_map pre-swizzled into WMMA B-fragment order (2 MB)
#define WS_GW 0
#define WS_A  2048
#define WS_C  2080
#define WS_SC 4096
#define WS_BP 266240

// ---------------------------------------------------------------------------
// Kernel 0: pre-swizzle W_map into per-lane B-fragment order so the GEMM can
// fetch 4 k-steps of B data with two global_load_b128 per 16x16 tile.
//   Bp[kg][n][hi][j][v] = W_map[kg*16 + 4*j + 2*hi + v][n]
// flat index = kg*16384 + n*16 + hi*8 + j*2 + v   (kg<32, n<1024, hi<2, j<4, v<2)
// ---------------------------------------------------------------------------
__global__ __launch_bounds__(256) void swizzle_kernel(
    const float* __restrict__ Wmap, float* __restrict__ ws)
{
    size_t o = (size_t)blockIdx.x * 256 + threadIdx.x;   // 524288 total
    int v  = (int)(o & 1);
    int j  = (int)((o >> 1) & 3);
    int hi = (int)((o >> 3) & 1);
    int n  = (int)((o >> 4) & 1023);
    int kg = (int)(o >> 14);
    int k  = kg * 16 + 4 * j + 2 * hi + v;
    ws[WS_BP + o] = Wmap[(size_t)k * DD + n];            // coalesced writes
}

// ---------------------------------------------------------------------------
// Kernel 1: batch-invariant precompute (1 block, 256 threads). Tiny work.
// ---------------------------------------------------------------------------
__global__ __launch_bounds__(256) void prep_kernel(
    const float* __restrict__ tt,  const float* __restrict__ ctk,
    const float* __restrict__ lng, const float* __restrict__ lnb,
    const float* __restrict__ Wq,  const float* __restrict__ bq,
    const float* __restrict__ Wk,  const float* __restrict__ bk,
    float* __restrict__ ws)
{
    __shared__ float hh[2][DD];
    __shared__ float qsh[2][HH][DHD];
    __shared__ float ksh[2][HH][DHD];
    __shared__ float wvs[2][DD];
    __shared__ float red[256];
    __shared__ float stat[2];
    const int tid = threadIdx.x;

    // LayerNorm of the two special rows (rows 0,1 of the concat sequence)
    for (int s = 0; s < 2; ++s) {
        const float* row = s ? ctk : tt;
        float lsum = 0.f;
        for (int i = tid; i < DD; i += 256) lsum += row[i];
        red[tid] = lsum; __syncthreads();
        for (int o = 128; o > 0; o >>= 1) { if (tid < o) red[tid] += red[tid + o]; __syncthreads(); }
        if (tid == 0) stat[0] = red[0] * (1.f / DD);
        __syncthreads();
        const float mu = stat[0];
        float lsq = 0.f;
        for (int i = tid; i < DD; i += 256) { float d = row[i] - mu; lsq += d * d; }
        __syncthreads();
        red[tid] = lsq; __syncthreads();
        for (int o = 128; o > 0; o >>= 1) { if (tid < o) red[tid] += red[tid + o]; __syncthreads(); }
        if (tid == 0) stat[1] = rsqrtf(red[0] * (1.f / DD) + EPSV);
        __syncthreads();
        const float rstd = stat[1];
        for (int i = tid; i < DD; i += 256)
            hh[s][i] = (row[i] - mu) * rstd * lng[i] + lnb[i];
        __syncthreads();
    }

    // q[s,h,e] = hh[s, h*64 : h*64+64] @ Wq[h] + bq[h]
    for (int j = tid; j < 2 * DD; j += 256) {
        int s = j >> 10, r = j & (DD - 1), h = r >> 6, e = r & 63;
        float acc = bq[h * DHD + e];
        const float* wq = Wq + h * DHD * DHD + e;
        const float* hr = &hh[s][h * DHD];
        for (int d = 0; d < DHD; ++d) acc += hr[d] * wq[d * DHD];
        qsh[s][h][e] = acc;
    }
    __syncthreads();

    // wv[s,d] = (Wk[h,dl,:] . q[s,h,:]) / 8 ;  gw = ln_g * wv
    for (int j = tid; j < 2 * DD; j += 256) {
        int s = j >> 10, d = j & (DD - 1), h = d >> 6, dl = d & 63;
        float wv = 0.f;
        const float* wk = Wk + h * DHD * DHD + dl * DHD;
        const float* q  = &qsh[s][h][0];
        for (int e = 0; e < DHD; ++e) wv += wk[e] * q[e];
        wv *= 0.125f;                       // fold 1/sqrt(DH)
        wvs[s][d] = wv;
        ws[WS_GW + s * DD + d] = lng[d] * wv;
    }
    // keys of the two special rows (needed for score columns t=0,1)
    for (int j = tid; j < 2 * DD; j += 256) {
        int t = j >> 10, r = j & (DD - 1), h = r >> 6, e = r & 63;
        float acc = bk[h * DHD + e];
        const float* wk = Wk + h * DHD * DHD + e;
        const float* hr = &hh[t][h * DHD];
        for (int d = 0; d < DHD; ++d) acc += hr[d] * wk[d * DHD];
        ksh[t][h][e] = acc;
    }
    __syncthreads();

    // Per-(s,h) folded constants (deterministic, fixed summation order)
    if (tid < 32) {
        int s = tid >> 4, h = tid & 15;
        float A = 0.f, C = 0.f;
        for (int dl = 0; dl < DHD; ++dl) {
            int d = h * DHD + dl;
            A += lng[d] * wvs[s][d];
            C += lnb[d] * wvs[s][d];
        }
        float ct = 0.f;
        for (int e = 0; e < DHD; ++e) ct += bk[h * DHD + e] * qsh[s][h][e];
        ws[WS_A + tid] = A;
        ws[WS_C + tid] = C + 0.125f * ct;
    }
    // Score columns t=0,1 (identical across batch)
    if (tid < 64) {
        int s = tid >> 5, rem = tid & 31, h = rem >> 1, t = rem & 1;
        float sc = 0.f;
        for (int e = 0; e < DHD; ++e) sc += qsh[s][h][e] * ksh[t][h][e];
        sc *= 0.125f;
        for (int b = 0; b < BD; ++b)
            ws[WS_SC + ((size_t)(b * 2 + s) * HH + h) * SSQ + t] = sc;
    }
}

// ---------------------------------------------------------------------------
// Kernel 2: fused  x@W_map (+bias)  ->  LN stats  ->  32 scores per token.
// One block = 16 tokens of one batch; 8 waves each own 128 output columns.
// GEMM via V_WMMA_F32_16X16X4_F32; B-fragments via b128 from pre-swizzled Bp.
// ---------------------------------------------------------------------------
#define XS 516   // LDS row stride: 516 % 64 == 4 -> banks 4*m + k, conflict-free

__global__ __launch_bounds__(256) void main_kernel(
    const float* __restrict__ x, const float* __restrict__ bmap,
    float* __restrict__ ws)
{
    __shared__ __attribute__((aligned(16))) float xs[16 * XS];
    __shared__ float rsP[16][8], rqP[16][8];
    __shared__ float Pl[512];               // [s][h][m]
    __shared__ float mus[16], rstds[16];
    const int tid  = threadIdx.x;
    const int b    = blockIdx.x >> 7;
    const int tile = blockIdx.x & 127;
    const int t0   = tile * 16;

    // Stage the 16x512 fp32 x-tile into LDS with b128 (tail rows zeroed)
    for (int j = tid; j < 16 * IND / 4; j += 256) {      // 2048 float4s
        int r = j >> 7, c4 = (j & 127) * 4;
        int gr = t0 + r;
        v4f val = {0.f, 0.f, 0.f, 0.f};
        if (gr < NN) val = *(const v4f*)(x + ((size_t)b * NN + gr) * IND + c4);
        *(v4f*)(xs + r * XS + c4) = val;
    }
    __syncthreads();

    const int wave = tid >> 5, lane = tid & 31;
    const int ml = lane & 15, hi = lane >> 4;

    // 8 accumulators = 16 rows x 128 cols per wave
    v8f acc[8] = {};
    const float* xrow = &xs[ml * XS];
    const float* Bp = ws + WS_BP;
    // per-lane base into swizzled B: kg*16384 + n*16 + hi*8
    const float* bpb = Bp + (size_t)(wave * 128 + ml) * 16 + hi * 8;

    for (int kg = 0; kg < 32; ++kg) {
        // Preload B-fragments for 4 k-steps: two b128 per tile
        v4f bf0[8], bf1[8];
        const float* bg = bpb + (size_t)kg * 16384;
#pragma unroll
        for (int c = 0; c < 8; ++c) {
            bf0[c] = *(const v4f*)(bg + c * 256);        // j=0,1 (v0,v1 pairs)
            bf1[c] = *(const v4f*)(bg + c * 256 + 4);    // j=2,3
        }
#pragma unroll
        for (int j = 0; j < 4; ++j) {
            const int k0 = kg * 16 + j * 4;
            // A 16x4 fp32 frag: lane(M=ml) holds K = k0 + 2*hi + {0,1}
            v2f a;
            a.x = xrow[k0 + 2 * hi];
            a.y = xrow[k0 + 2 * hi + 1];
#pragma unroll
            for (int c = 0; c < 8; ++c) {
                v2f bv;
                if (j < 2) { bv.x = bf0[c][j * 2];       bv.y = bf0[c][j * 2 + 1]; }
                else       { bv.x = bf1[c][(j - 2) * 2]; bv.y = bf1[c][(j - 2) * 2 + 1]; }
                acc[c] = __builtin_amdgcn_wmma_f32_16x16x4_f32(
                    false, a, false, bv, (short)0, acc[c], false, false);
            }
        }
    }

    // Fused bias + row stats + segmented dots with gw[0/1]
    float srow[8], sq[8], p0[2][8], p1[2][8];
#pragma unroll
    for (int r = 0; r < 8; ++r) {
        srow[r] = 0.f; sq[r] = 0.f;
        p0[0][r] = 0.f; p0[1][r] = 0.f; p1[0][r] = 0.f; p1[1][r] = 0.f;
    }
    const float* gw = ws + WS_GW;
#pragma unroll
    for (int c = 0; c < 8; ++c) {
        int n = wave * 128 + c * 16 + ml;
        float bm = bmap[n];
        float g0 = gw[n], g1 = gw[DD + n];
        const int hl = c >> 2;              // wave covers heads 2w, 2w+1
#pragma unroll
        for (int r = 0; r < 8; ++r) {
            float v = acc[c][r] + bm;       // C/D layout: row m = r + 8*hi, col n
            srow[r] += v; sq[r] += v * v;
            p0[hl][r] += v * g0;
            p1[hl][r] += v * g1;
        }
    }
    // Reduce across the 16 lanes sharing a row group (xor<16 stays in half-wave)
#pragma unroll
    for (int m = 1; m < 16; m <<= 1) {
#pragma unroll
        for (int r = 0; r < 8; ++r) {
            srow[r]  += __shfl_xor(srow[r],  m, 32);
            sq[r]    += __shfl_xor(sq[r],    m, 32);
            p0[0][r] += __shfl_xor(p0[0][r], m, 32);
            p0[1][r] += __shfl_xor(p0[1][r], m, 32);
            p1[0][r] += __shfl_xor(p1[0][r], m, 32);
            p1[1][r] += __shfl_xor(p1[1][r], m, 32);
        }
    }
    if (ml == 0) {                           // lanes 0 (rows 0-7) and 16 (rows 8-15)
#pragma unroll
        for (int r = 0; r < 8; ++r) {
            int mrow = r + 8 * hi;
            rsP[mrow][wave] = srow[r];
            rqP[mrow][wave] = sq[r];
#pragma unroll
            for (int hl = 0; hl < 2; ++hl) {
                int h = 2 * wave + hl;
                Pl[0 * 256 + h * 16 + mrow] = (hl == 0) ? p0[0][r] : p0[1][r];
                Pl[1 * 256 + h * 16 + mrow] = (hl == 0) ? p1[0][r] : p1[1][r];
            }
        }
    }
    __syncthreads();
    if (tid < 16) {                          // deterministic cross-wave sum
        float s = 0.f, q = 0.f;
#pragma unroll
        for (int w = 0; w < 8; ++w) { s += rsP[tid][w]; q += rqP[tid][w]; }
        float mu = s * (1.f / DD);
        mus[tid] = mu;
        rstds[tid] = rsqrtf(q * (1.f / DD) - mu * mu + EPSV);
    }
    __syncthreads();

    // score[b,s,h,t] = rstd*(P - mu*A) + C
    const float* Aw = ws + WS_A;
    const float* Cw = ws + WS_C;
    float* sc = ws + WS_SC;
    for (int j = tid; j < 512; j += 256) {
        int s = j >> 8, rem = j & 255, h = rem >> 4, mrow = rem & 15;
        int tok = t0 + mrow;
        if (tok < NN) {
            float v = rstds[mrow] * (Pl[j] - mus[mrow] * Aw[s * 16 + h]) + Cw[s * 16 + h];
            sc[((size_t)(b * 2 + s) * HH + h) * SSQ + (tok + 2)] = v;
        }
    }
}

// ---------------------------------------------------------------------------
// Kernel 3: per-(b,s): softmax over t for each head, mean over heads.
// ---------------------------------------------------------------------------
__global__ __launch_bounds__(256) void finalize_kernel(
    const float* __restrict__ ws, float* __restrict__ out)
{
    __shared__ float buf[HH][256];
    __shared__ float gmax[HH], ginv[HH];
    const int tid = threadIdx.x;
    const int bs = blockIdx.x;               // b*2 + s
    const int b = bs >> 1, s = bs & 1;
    const float* sc = ws + WS_SC + (size_t)bs * HH * SSQ;

    float mx[HH];
#pragma unroll
    for (int h = 0; h < HH; ++h) mx[h] = -3.4e38f;
    for (int t = tid; t < SSQ; t += 256) {
#pragma unroll
        for (int h = 0; h < HH; ++h) mx[h] = fmaxf(mx[h], sc[h * SSQ + t]);
    }
#pragma unroll
    for (int h = 0; h < HH; ++h) buf[h][tid] = mx[h];
    __syncthreads();
    if (tid < HH) {
        float m = -3.4e38f;
        for (int i = 0; i < 256; ++i) m = fmaxf(m, buf[tid][i]);
        gmax[tid] = m;
    }
    __syncthreads();

    float se[HH];
#pragma unroll
    for (int h = 0; h < HH; ++h) se[h] = 0.f;
    for (int t = tid; t < SSQ; t += 256) {
#pragma unroll
        for (int h = 0; h < HH; ++h) se[h] += __expf(sc[h * SSQ + t] - gmax[h]);
    }
    __syncthreads();
#pragma unroll
    for (int h = 0; h < HH; ++h) buf[h][tid] = se[h];
    __syncthreads();
    if (tid < HH) {
        float z = 0.f;
        for (int i = 0; i < 256; ++i) z += buf[tid][i];
        ginv[tid] = 1.f / z;
    }
    __syncthreads();

    for (int t = tid; t < SSQ; t += 256) {
        float a = 0.f;
#pragma unroll
        for (int h = 0; h < HH; ++h)
            a += __expf(sc[h * SSQ + t] - gmax[h]) * ginv[h];
        out[(size_t)s * BD * SSQ + (size_t)b * SSQ + t] = a * (1.f / HH);
    }
}

// ---------------------------------------------------------------------------
extern "C" void kernel_launch(void* const* d_in, const int* in_sizes, int n_in,
                              void* d_out, int out_size, void* d_ws, size_t ws_size,
                              hipStream_t stream)
{
    const float* x    = (const float*)d_in[0];
    const float* Wmap = (const float*)d_in[1];
    const float* bmap = (const float*)d_in[2];
    const float* tt   = (const float*)d_in[3];
    const float* ctk  = (const float*)d_in[4];
    const float* lng  = (const float*)d_in[5];
    const float* lnb  = (const float*)d_in[6];
    const float* Wq   = (const float*)d_in[7];
    const float* bq   = (const float*)d_in[8];
    const float* Wk   = (const float*)d_in[9];
    const float* bk   = (const float*)d_in[10];
    float* ws  = (float*)d_ws;
    float* out = (float*)d_out;

    swizzle_kernel<<<dim3(2048), dim3(256), 0, stream>>>(Wmap, ws);
    prep_kernel<<<dim3(1), dim3(256), 0, stream>>>(tt, ctk, lng, lnb, Wq, bq, Wk, bk, ws);
    main_kernel<<<dim3(BD * 128), dim3(256), 0, stream>>>(x, bmap, ws);
    finalize_kernel<<<dim3(BD * 2), dim3(256), 0, stream>>>(ws, out);
}